// FeedbackAttention_67851893342654
// MI455X (gfx1250) — compile-verified
//
#include <hip/hip_runtime.h>
#include <math.h>

// FeedbackAttention, algebraically folded to the memory-roofline form.
// B=8, S=4096, H=16, DK=128, D=2048.
// k3_scores / k5_uval stream key/value exactly once (~640 MB total ->
// ~27us at 23.3 TB/s). k3 now stages tiles with the Tensor Data Mover
// (tensor_load_to_lds, TENSORcnt double-buffered) and both use
// V_WMMA_F32_16X16X4_F32 (full f32 precision, f32 accumulate).

#define BB 8
#define SS 4096
#define DD 2048
#define HH 16
#define DKK 128
#define BD (BB*DD)   // 16384
#define BHh (BB*HH)  // 128
#define SCALE 0.08838834764831845f  // 1/sqrt(128)

typedef __attribute__((ext_vector_type(2))) float v2f;
typedef __attribute__((ext_vector_type(8))) float v8f;
typedef __attribute__((ext_vector_type(4))) unsigned int v4u;
typedef __attribute__((ext_vector_type(4))) int v4i;
typedef __attribute__((ext_vector_type(8))) int v8i;

// Toolchain probe: the 6-arg tensor_load_to_lds builtin ships with the
// toolchain that also provides the gfx1250 TDM header; ROCm 7.2 is 5-arg.
#if __has_include(<hip/amd_detail/amd_gfx1250_TDM.h>)
#define TDM_SIX_ARGS 1
#else
#define TDM_SIX_ARGS 0
#endif

// Issue one TDM 2D tile load: tile_d0 contiguous f32 per row, tile_d1 rows,
// row stride stride_el (f32 units), LDS padding +2 dwords every 64 dwords
// (so a 64-wide row lands at LDS stride 66 -> bank-conflict-free WMMA reads).
__device__ __forceinline__ void tdm_load_tile(unsigned int lds_off,
                                              const float* gptr,
                                              unsigned int tile_d0,
                                              unsigned int tile_d1,
                                              unsigned long long stride_el) {
  unsigned long long ga = (unsigned long long)(uintptr_t)gptr;
  v4u g0;
  g0[0] = 1u;                                   // count=1 valid descriptor
  g0[1] = lds_off;                              // lds_addr (bytes)
  g0[2] = (unsigned int)(ga & 0xffffffffull);   // global_addr[31:0]
  g0[3] = (unsigned int)((ga >> 32) & 0x1ffffffull) | (2u << 30); // type=2
  unsigned int td0 = 1u << 16, td1 = 1u << 20;  // tensor dims (no OOB use)
  v8i g1;
  g1[0] = (int)((2u << 16)        // data_size = 4 bytes
              | (1u << 20)        // pad_enable
              | (5u << 22)        // pad_interval code 5 = every 64 dwords
              | (1u << 25));      // pad_amount  code 1 = 2 dwords
  g1[1] = (int)((td0 & 0xffffu) << 16);                       // tensor_dim0 lo
  g1[2] = (int)(((td0 >> 16) & 0xffffu) | ((td1 & 0xffffu) << 16));
  g1[3] = (int)(((td1 >> 16) & 0xffffu) | (tile_d0 << 16));   // tile_dim0
  g1[4] = (int)(tile_d1 & 0xffffu);                           // tile_dim1
  g1[5] = (int)(unsigned int)(stride_el & 0xffffffffull);     // dim0_stride lo
  g1[6] = (int)(unsigned int)((stride_el >> 32) & 0xffffull); // dim0_stride hi
  g1[7] = 0;                                                  // dim1_stride=0
  v4i z4 = {0, 0, 0, 0};
#if TDM_SIX_ARGS
  v8i z8 = {0, 0, 0, 0, 0, 0, 0, 0};
  __builtin_amdgcn_tensor_load_to_lds(g0, g1, z4, z4, z8, 0);
#else
  __builtin_amdgcn_tensor_load_to_lds(g0, g1, z4, z4, 0);
#endif
}

// ---------------- K1: qb[b,n] = query @ Wq + q_bias (tiny: 67 MFLOP) -------
__global__ void k1_qproj(const float* __restrict__ query, const float* __restrict__ Wq,
                         const float* __restrict__ q_bias, float* __restrict__ qb) {
  int t = blockIdx.x * blockDim.x + threadIdx.x;   // 16384 threads
  int b = t / DD, n = t % DD;
  float acc = 0.f;
  for (int k = 0; k < DD; ++k) acc += query[b*DD + k] * Wq[(size_t)k*DD + n];
  qb[t] = acc + q_bias[n];
}

// ------- K2: wq_k[b,h,kk] = sum_d Wk[kk, h*128+d] * qb[b, h*128+d] ---------
__global__ void k2_fold_wk(const float* __restrict__ Wk, const float* __restrict__ qb,
                           float* __restrict__ wq_k) {
  int t = blockIdx.x * blockDim.x + threadIdx.x;   // 262144 threads
  int b  = t >> 15;
  int r  = t & 32767;
  int h  = r >> 11;
  int kk = r & 2047;
  const float4* wrow = (const float4*)&Wk[(size_t)kk*DD + h*DKK];
  const float4* qrow = (const float4*)&qb[b*DD + h*DKK];
  float acc = 0.f;
  #pragma unroll 8
  for (int i = 0; i < DKK/4; ++i) {
    float4 w = wrow[i], q = qrow[i];
    acc += w.x*q.x + w.y*q.y + w.z*q.z + w.w*q.w;
  }
  wq_k[t] = acc;                                   // layout [b][h][kk] == t
}

// ---- K3: scores[j,b,h] = SCALE * key[j,b,:] . wq_k[b,h,:]   (streams key) --
// One block (4 waves) per (b, 64-row j group). Wave 0 drives the Tensor Data
// Mover: per 64-wide K chunk, TDM DMAs a 64x64 key tile (row stride B*D) and
// a 16x64 wq_k tile straight into LDS (padded to stride 66), double-buffered
// on TENSORcnt. Each wave computes one 16x16 tile with 16 f32 WMMAs/chunk.
#define KC 64
#define LSTP 66                   // LDS row stride after TDM padding
#define A_TILE (64*LSTP + 2)      // + slack for trailing pad
#define B_TILE (16*LSTP + 2)
__global__ void __launch_bounds__(128)
k3_scores(const float* __restrict__ key, const float* __restrict__ wq_k,
          float* __restrict__ scores) {
  __shared__ float As[2][A_TILE];
  __shared__ float Bs[2][B_TILE];
  int tid  = threadIdx.x;
  int lane = tid & 31;
  int w    = tid >> 5;
  int b    = blockIdx.x >> 6;                    // 8 b
  int j0   = (blockIdx.x & 63) << 6;             // 64 j-groups of 64 rows
  int j0w  = j0 + w * 16;                        // this wave's 16 rows
  int half = lane >> 4, m15 = lane & 15;

  const float* keyp = key  + (size_t)j0 * BD + (size_t)b * DD;
  const float* wqp  = wq_k + (size_t)b * HH * DD;
  unsigned int ldsA0 = (unsigned int)(uintptr_t)&As[0][0];
  unsigned int ldsA1 = (unsigned int)(uintptr_t)&As[1][0];
  unsigned int ldsB0 = (unsigned int)(uintptr_t)&Bs[0][0];
  unsigned int ldsB1 = (unsigned int)(uintptr_t)&Bs[1][0];

  if (tid < 32) {                                // wave 0 preloads chunk 0
    tdm_load_tile(ldsA0, keyp + 0, KC, 64, (unsigned long long)BD);
    tdm_load_tile(ldsB0, wqp  + 0, KC, 16, (unsigned long long)DD);
  }

  v8f c = {};
  int buf = 0;
  const int NCHUNK = DD / KC;                    // 32
  for (int cidx = 0; cidx < NCHUNK; ++cidx) {
    __syncthreads();   // all waves done reading the buffer TDM will refill
    if (tid < 32) {
      if (cidx + 1 < NCHUNK) {
        int k0n = (cidx + 1) * KC;
        tdm_load_tile(buf ? ldsA0 : ldsA1, keyp + k0n, KC, 64,
                      (unsigned long long)BD);
        tdm_load_tile(buf ? ldsB0 : ldsB1, wqp + k0n, KC, 16,
                      (unsigned long long)DD);
        __builtin_amdgcn_s_wait_tensorcnt(2);    // current pair complete
      } else {
        __builtin_amdgcn_s_wait_tensorcnt(0);    // last chunk: drain
      }
    }
    __syncthreads();   // data for `buf` visible to all waves
    const float* a_s = As[buf];
    const float* b_s = Bs[buf];
    #pragma unroll
    for (int kk = 0; kk < KC; kk += 4) {
      v2f a  = *(const v2f*)&a_s[(w*16 + m15)*LSTP + kk + half*2];
      v2f bb = *(const v2f*)&b_s[m15*LSTP + kk + half*2];
      c = __builtin_amdgcn_wmma_f32_16x16x4_f32(false, a, false, bb,
                                                (short)0, c, false, false);
    }
    buf ^= 1;
  }
  #pragma unroll
  for (int i = 0; i < 8; ++i) {
    int M = i + half*8;
    scores[(size_t)(j0w + M)*BHh + b*HH + m15] = c[i] * SCALE;
  }
}

// ---- K3b: scores[j,b,h] += SCALE * qb[b,h,:] . key_pos[j,h,:]  (34 MB) ----
__global__ void k3b_pos(const float* __restrict__ key_pos, const float* __restrict__ qb,
                        float* __restrict__ scores) {
  int lane = threadIdx.x & 31;
  int w    = threadIdx.x >> 5;
  int gw   = blockIdx.x * 8 + w;                   // 65536 waves: one per (j,h)
  int j = gw >> 4, h = gw & 15;
  float4 kp = *(const float4*)&key_pos[(size_t)(j*HH + h)*DKK + lane*4];
  float s[8];
  #pragma unroll
  for (int b = 0; b < BB; ++b) {
    float4 q = *(const float4*)&qb[b*DD + h*DKK + lane*4];
    float p = kp.x*q.x + kp.y*q.y + kp.z*q.z + kp.w*q.w;
    #pragma unroll
    for (int off = 16; off > 0; off >>= 1) p += __shfl_xor(p, off, 32);
    s[b] = p;
  }
  if (lane == 0) {
    #pragma unroll
    for (int b = 0; b < BB; ++b)
      scores[(size_t)j*BHh + b*HH + h] += SCALE * s[b];
  }
}

// ---------------- K4: softmax over j for each (b,h) column, in place -------
__global__ void k4_softmax(float* __restrict__ scores) {
  __shared__ float red[256];
  int col = blockIdx.x;                            // 128 columns = b*16+h
  int tid = threadIdx.x;
  float m = -1e30f;
  for (int j = tid; j < SS; j += 256) m = fmaxf(m, scores[(size_t)j*BHh + col]);
  red[tid] = m; __syncthreads();
  for (int o = 128; o > 0; o >>= 1) {
    if (tid < o) red[tid] = fmaxf(red[tid], red[tid+o]);
    __syncthreads();
  }
  m = red[0]; __syncthreads();
  float sum = 0.f;
  for (int j = tid; j < SS; j += 256) sum += __expf(scores[(size_t)j*BHh + col] - m);
  red[tid] = sum; __syncthreads();
  for (int o = 128; o > 0; o >>= 1) {
    if (tid < o) red[tid] += red[tid+o];
    __syncthreads();
  }
  float inv = 1.f / red[0];
  for (int j = tid; j < SS; j += 256) {
    size_t idx = (size_t)j*BHh + col;
    scores[idx] = __expf(scores[idx] - m) * inv;
  }
}

// ---- K5: u[b,h,d] = sum_j attn[j,b,h] * value[j,b,d]   (streams value) ----
// One wave per (b, 16-col d tile). C is 16(heads) x 16(d), K runs over j.
#define K5_WPB 4
__global__ void __launch_bounds__(128)
k5_uval(const float* __restrict__ attn, const float* __restrict__ value,
        float* __restrict__ u) {
  int lane = threadIdx.x & 31;
  int w    = threadIdx.x >> 5;
  int t    = blockIdx.x * K5_WPB + w;              // 1024 waves
  int b    = t >> 7;
  int d0   = (t & 127) << 4;
  int half = lane >> 4, m15 = lane & 15;
  const float* ap = attn  + b*HH + m15;
  const float* vp = value + (size_t)b*DD + d0 + m15;
  v8f c = {};
  #pragma unroll 4
  for (int j = 0; j < SS; j += 4) {
    int r0 = j + half*2, r1 = r0 + 1;
    v2f a, bb;
    a.x  = ap[(size_t)r0*BHh]; a.y  = ap[(size_t)r1*BHh];
    bb.x = vp[(size_t)r0*BD];  bb.y = vp[(size_t)r1*BD];
    c = __builtin_amdgcn_wmma_f32_16x16x4_f32(false, a, false, bb,
                                              (short)0, c, false, false);
  }
  #pragma unroll
  for (int i = 0; i < 8; ++i) {
    int hM = i + half*8;
    u[(size_t)b*HH*DD + (size_t)hM*DD + d0 + m15] = c[i];
  }
}

// -------- K6a: x[b, h*128+dd] = u[b,h,:] @ Wv[:, h*128+dd] + bv ------------
__global__ void k6a_xproj(const float* __restrict__ u, const float* __restrict__ Wv,
                          const float* __restrict__ bv, float* __restrict__ x) {
  int t = blockIdx.x * blockDim.x + threadIdx.x;   // 16384 threads
  int b = t / DD, n = t % DD;
  int h = n / DKK;
  const float* up = u + (size_t)b*HH*DD + (size_t)h*DD;
  float acc = 0.f;
  for (int d = 0; d < DD; ++d) acc += up[d] * Wv[(size_t)d*DD + n];
  x[t] = acc + bv[n];
}

// -------- K6b: out[b,n] = x[b,:] @ Wo[:,n] + bo ----------------------------
__global__ void k6b_oproj(const float* __restrict__ x, const float* __restrict__ Wo,
                          const float* __restrict__ bo, float* __restrict__ out) {
  int t = blockIdx.x * blockDim.x + threadIdx.x;   // 16384 threads
  int b = t / DD, n = t % DD;
  float acc = 0.f;
  for (int m = 0; m < DD; ++m) acc += x[b*DD + m] * Wo[(size_t)m*DD + n];
  out[t] = acc + bo[n];
}

extern "C" void kernel_launch(void* const* d_in, const int* in_sizes, int n_in,
                              void* d_out, int out_size, void* d_ws, size_t ws_size,
                              hipStream_t stream) {
  const float* query   = (const float*)d_in[0];
  const float* key     = (const float*)d_in[1];
  const float* value   = (const float*)d_in[2];
  const float* Wq      = (const float*)d_in[3];
  const float* Wk      = (const float*)d_in[4];
  const float* Wv      = (const float*)d_in[5];
  const float* bv      = (const float*)d_in[6];
  const float* Wo      = (const float*)d_in[7];
  const float* bo      = (const float*)d_in[8];
  const float* key_pos = (const float*)d_in[9];
  const float* q_bias  = (const float*)d_in[10];
  float* out = (float*)d_out;

  // workspace layout (floats)
  float* ws     = (float*)d_ws;
  float* qb     = ws;                        // 16384
  float* wq_k   = qb + 16384;                // 262144
  float* scores = wq_k + 262144;             // 524288 (becomes attn in place)
  float* u      = scores + 524288;           // 262144
  float* x      = u + 262144;                // 16384   (total ~4.3 MB)

  k1_qproj  <<<  64, 256, 0, stream>>>(query, Wq, q_bias, qb);
  k2_fold_wk<<<1024, 256, 0, stream>>>(Wk, qb, wq_k);
  k3_scores <<< 512, 128, 0, stream>>>(key, wq_k, scores);
  k3b_pos   <<<8192, 256, 0, stream>>>(key_pos, qb, scores);
  k4_softmax<<< 128, 256, 0, stream>>>(scores);
  k5_uval   <<< 256, 128, 0, stream>>>(scores, value, u);
  k6a_xproj <<<  64, 256, 0, stream>>>(u, Wv, bv, x);
  k6b_oproj <<<  64, 256, 0, stream>>>(x, Wo, bo, out);
}